// RadiusFilter_21878563405909
// MI455X (gfx1250) — compile-verified
//
#include <hip/hip_runtime.h>

typedef float v2f __attribute__((ext_vector_type(2)));
typedef float v8f __attribute__((ext_vector_type(8)));

static constexpr int LPTS = 8192;   // points per batch
static constexpr int NB   = 4;      // batches
static constexpr float R2 = 0.05f * 0.05f;
static constexpr int NNEI = 8;

// One wave (32 lanes) computes a 16-row tile of the count vector.
// 8 waves per block; block stages the whole batch's x/y into LDS (64 KB).
//
// d_ij = s_i + s_j - 2*G_ij with G = P P^T (K=2, zero-padded to K=4 for
// v_wmma_f32_16x16x4_f32).  Count condition rearranged to
//   G_ij >= ti + tj,   ti = (s_i - R^2)/2,  tj = s_j/2
// so the per-tile epilogue is one pk_add (threshold) + compare per 2 elems.
__global__ __launch_bounds__(256) void radius_count_wmma(
    const float* __restrict__ x, const float* __restrict__ y,
    const unsigned char* __restrict__ valid, float* __restrict__ mask_out)
{
    __shared__ float sx[LPTS];
    __shared__ float sy[LPTS];

    const int b = blockIdx.y;
    const float* xb = x + b * LPTS;
    const float* yb = y + b * LPTS;

    // Cooperative stage of the full batch coordinates into LDS.
    for (int i = threadIdx.x; i < LPTS; i += 256) {
        sx[i] = xb[i];
        sy[i] = yb[i];
    }
    __syncthreads();

    const int wave = threadIdx.x >> 5;        // 0..7
    const int lane = threadIdx.x & 31;
    const int hi   = lane >> 4;               // 0: lanes 0-15, 1: lanes 16-31
    const int l16  = lane & 15;
    const int row0 = (blockIdx.x * 8 + wave) * 16;

    // A (16x4 f32): lanes 0-15 carry K=0 (x_i) and K=1 (y_i); lanes 16-31
    // hold the zero padding for K=2,3.
    v2f A;
    A.x = hi ? 0.0f : sx[row0 + l16];
    A.y = hi ? 0.0f : sy[row0 + l16];

    // Per-lane row thresholds ti[v] = (s_i - R^2) * 0.5 for the 8 rows this
    // lane's C registers correspond to.
    float ti[8];
#pragma unroll
    for (int v = 0; v < 8; ++v) {
        const int r = row0 + v + 8 * hi;
        const float xv = sx[r], yv = sy[r];
        ti[v] = (xv * xv + yv * yv - R2) * 0.5f;
    }

    int cnt[8];
#pragma unroll
    for (int v = 0; v < 8; ++v) cnt[v] = 0;

    for (int ct = 0; ct < LPTS / 16; ++ct) {
        const int   jj = ct * 16 + l16;
        const float xj = sx[jj];
        const float yj = sy[jj];
        const float tj = (xj * xj + yj * yj) * 0.5f;   // s_j / 2

        // B (4x16 f32): lanes 0-15 carry rows K=0,1; lanes 16-31 carry the
        // zero-padded rows K=2,3.
        v2f Bm;
        Bm.x = hi ? 0.0f : xj;
        Bm.y = hi ? 0.0f : yj;

        v8f C = {};
        // G tile = A x B (+0): v_wmma_f32_16x16x4_f32
        C = __builtin_amdgcn_wmma_f32_16x16x4_f32(
                /*neg_a=*/false, A, /*neg_b=*/false, Bm,
                /*c_mod=*/(short)0, C, /*reuse_a=*/false, /*reuse_b=*/false);

#pragma unroll
        for (int v = 0; v < 8; ++v) {
            // d <= R^2  <=>  G >= ti + tj
            cnt[v] += (C[v] >= ti[v] + tj) ? 1 : 0;
        }
    }

    // Reduce each row counter across its 16-lane half (wave32 shuffles).
#pragma unroll
    for (int v = 0; v < 8; ++v) {
        int c = cnt[v];
        c += __shfl_xor(c, 1, 32);
        c += __shfl_xor(c, 2, 32);
        c += __shfl_xor(c, 4, 32);
        c += __shfl_xor(c, 8, 32);
        cnt[v] = c;
    }

    if (l16 == 0) {
#pragma unroll
        for (int v = 0; v < 8; ++v) {
            const int r = row0 + v + 8 * hi;
            const bool m = (cnt[v] > NNEI) && (valid[b * LPTS + r] != 0);
            mask_out[b * LPTS + r] = m ? 1.0f : 0.0f;
        }
    }
}

// Pass-through copies of p, y, x, t into the head of d_out.
__global__ __launch_bounds__(256) void copy4_kernel(
    const float* __restrict__ p, const float* __restrict__ y,
    const float* __restrict__ x, const float* __restrict__ t,
    float* __restrict__ out)
{
    const int idx   = blockIdx.x * 256 + threadIdx.x;   // 0 .. 4*NB*LPTS-1
    const int which = idx >> 15;                         // / (NB*LPTS)
    const int off   = idx & (NB * LPTS - 1);
    const float* src = (which == 0) ? p : (which == 1) ? y : (which == 2) ? x : t;
    out[idx] = src[off];
}

extern "C" void kernel_launch(void* const* d_in, const int* in_sizes, int n_in,
                              void* d_out, int out_size, void* d_ws, size_t ws_size,
                              hipStream_t stream)
{
    const float* p = (const float*)d_in[0];
    const float* y = (const float*)d_in[1];
    const float* x = (const float*)d_in[2];
    const float* t = (const float*)d_in[3];
    const unsigned char* valid = (const unsigned char*)d_in[4];

    float* out      = (float*)d_out;
    float* mask_out = out + 4 * NB * LPTS;

    // Copies: 4 * 4 * 8192 = 131072 elements.
    copy4_kernel<<<(4 * NB * LPTS) / 256, 256, 0, stream>>>(p, y, x, t, out);

    // Counting: 512 row tiles / 8 waves per block = 64 blocks in x, 4 batches in y.
    dim3 grid(LPTS / 16 / 8, NB);
    radius_count_wmma<<<grid, 256, 0, stream>>>(x, y, valid, mask_out);
}